// GNNLayer_10075993277103
// MI455X (gfx1250) — compile-verified
//
#include <hip/hip_runtime.h>
#include <math.h>

typedef float v2f __attribute__((ext_vector_type(2)));
typedef float v8f __attribute__((ext_vector_type(8)));

#define B_    64
#define T_    12
#define N_    207
#define H_    64
#define TW_   10    // T-2 output windows
#define NW_   621   // 3*N window nodes (contiguous in memory!)
#define ROW0_ 414   // first row of last-timestep block (2*N)
#define NTILE_ 13   // ceil(207/16) output row tiles
#define MTILES_ 39  // ceil(621/16) window column tiles

struct TrueT  { static constexpr bool value = true;  };
struct FalseT { static constexpr bool value = false; };

// D(16x16,f32) = A(16x4,f32) * B(4x16,f32) + C  -- full fp32 on the matrix pipe
static __device__ __forceinline__ v8f wmma_f32(v2f a, v2f b, v8f c) {
  return __builtin_amdgcn_wmma_f32_16x16x4_f32(false, a, false, b, (short)0, c,
                                               false, false);
}

// ---------------------------------------------------------------------------
// Pass 1: wf = L2-normalize(feat * sigmoid(weights)) per node, into workspace.
// ---------------------------------------------------------------------------
__global__ void gnn_prep_kernel(const float* __restrict__ feat,
                                const float* __restrict__ w,
                                float* __restrict__ wf) {
  int p = blockIdx.x * blockDim.x + threadIdx.x;
  if (p >= B_ * T_ * N_) return;
  const float* src = feat + (size_t)p * H_;
  float* dst = wf + (size_t)p * H_;
  float tmp[H_];
  float ss = 0.f;
#pragma unroll
  for (int i = 0; i < H_; ++i) {
    float sw = 1.f / (1.f + __expf(-w[i]));
    float v = src[i] * sw;
    tmp[i] = v;
    ss += v * v;
  }
  float scale = 1.f / fmaxf(sqrtf(ss), 1e-12f);
#pragma unroll
  for (int i = 0; i < H_; ++i) dst[i] = tmp[i] * scale;
}

// ---------------------------------------------------------------------------
// Pass 2: per-wave 16-row output tile of one (b, t) window.
//   S = winw_rows(16x64) . winw_tile(16x64)^T       (WMMA, identical A/B load)
//   P = relu(S); rowsums via WMMA P @ ones          (no cross-lane shuffles)
//   agg += P(16x16) @ win_tile(16x64)               (WMMA, P relayout via LDS)
//   agg *= 1/rowsum; FFN (2x WMMA chains); residual; LayerNorm; store.
//   Tail m-tile (partial 13 rows) is peeled so the 38 full iterations carry
//   no masking/clamping VALU at all.
// ---------------------------------------------------------------------------
__global__ __launch_bounds__(256, 1) void gnn_main_kernel(
    const float* __restrict__ feat, const float* __restrict__ wf,
    const float* __restrict__ W1, const float* __restrict__ b1,
    const float* __restrict__ W2, const float* __restrict__ b2,
    const float* __restrict__ gamma, const float* __restrict__ beta,
    float* __restrict__ out) {
  __shared__ float lds[8][16 * 68];  // per-wave 16x64 staging tile (+pad)
  const int lane = threadIdx.x & 31;
  const int wid  = threadIdx.x >> 5;
  const int gw   = blockIdx.x * 8 + wid;
  if (gw >= B_ * TW_ * NTILE_) return;
  const int tile = gw % NTILE_;
  const int bt   = gw / NTILE_;
  const int t    = bt % TW_;
  const int b    = bt / TW_;
  const int rowM = lane & 15;
  const int hi   = lane >> 4;
  float* myl = lds[wid];

  const float* win_base = feat + (size_t)((b * T_ + t) * N_) * H_;  // 621x64
  const float* wf_base  = wf   + (size_t)((b * T_ + t) * N_) * H_;  // 621x64

  // A operand: the wave's 16 output rows (clamped duplicates for the tail tile
  // produce garbage rows that are simply never written out).
  int arow = ROW0_ + tile * 16 + rowM;
  if (arow > NW_ - 1) arow = NW_ - 1;
  v2f a[16];
#pragma unroll
  for (int k = 0; k < 16; ++k)
    a[k] = *(const v2f*)(wf_base + (size_t)arow * H_ + k * 4 + 2 * hi);

  v8f agg[4];
  v8f rsacc = {0.f, 0.f, 0.f, 0.f, 0.f, 0.f, 0.f, 0.f};
#pragma unroll
  for (int hb = 0; hb < 4; ++hb)
#pragma unroll
    for (int r = 0; r < 8; ++r) agg[hb][r] = 0.f;

  v2f onesb;
  onesb[0] = 1.f;
  onesb[1] = 1.f;

  auto mtile = [&](int mt, auto tailc) {
    constexpr bool TAIL = decltype(tailc)::value;
    const int mrowbase = mt * 16;
    bool mv = true;
    int mrow = mrowbase + rowM;
    if (TAIL) {
      mv = (mrow < NW_);
      if (!mv) mrow = NW_ - 1;
    }
    const float* brow = wf_base + (size_t)mrow * H_;
    if (!TAIL) {  // global_prefetch_b8 the next tile's rows
      int pr = mrowbase + 16 + rowM;
      if (pr > NW_ - 1) pr = NW_ - 1;
      __builtin_prefetch(wf_base + (size_t)pr * H_ + 2 * hi, 0, 3);
    }
    // ---- S = A . Y^T : B operand loads with the SAME layout as A ----
    v8f S = {0.f, 0.f, 0.f, 0.f, 0.f, 0.f, 0.f, 0.f};
#pragma unroll
    for (int k = 0; k < 16; ++k) {
      v2f bop = *(const v2f*)(brow + k * 4 + 2 * hi);
      if (TAIL && !mv) {  // zero => invalid columns contribute exactly 0
        bop[0] = 0.f;
        bop[1] = 0.f;
      }
      S = wmma_f32(a[k], bop, S);
    }
    // ---- P = relu(S), staged row-major in LDS for A-layout re-read ----
#pragma unroll
    for (int r = 0; r < 8; ++r)
      myl[(r + 8 * hi) * 68 + rowM] = fmaxf(S[r], 0.f);
    v2f pa[4];
#pragma unroll
    for (int j = 0; j < 4; ++j)
      pa[j] = *(const v2f*)(myl + rowM * 68 + j * 4 + 2 * hi);
    // ---- row sums on the matrix pipe: rsacc += P @ ones ----
#pragma unroll
    for (int j = 0; j < 4; ++j) rsacc = wmma_f32(pa[j], onesb, rsacc);
    // ---- agg += P(16x16) @ V(16x64) ----
#pragma unroll
    for (int hb = 0; hb < 4; ++hb) {
#pragma unroll
      for (int j = 0; j < 4; ++j) {
        int r0 = mrowbase + j * 4 + 2 * hi;
        int r1 = r0 + 1;
        if (TAIL) {  // clamped rows multiply P==0 => exact
          if (r0 > NW_ - 1) r0 = NW_ - 1;
          if (r1 > NW_ - 1) r1 = NW_ - 1;
        }
        v2f vb;
        vb[0] = win_base[(size_t)r0 * H_ + hb * 16 + rowM];
        vb[1] = win_base[(size_t)r1 * H_ + hb * 16 + rowM];
        agg[hb] = wmma_f32(pa[j], vb, agg[hb]);
      }
    }
  };

  for (int mt = 0; mt < MTILES_ - 1; ++mt) mtile(mt, FalseT{});
  mtile(MTILES_ - 1, TrueT{});

  // ---- deferred D^-1 row normalization: rsacc reg r == rowsum of agg reg r
#pragma unroll
  for (int r = 0; r < 8; ++r) {
    float inv = 1.f / fmaxf(rsacc[r], 1e-12f);
#pragma unroll
    for (int hb = 0; hb < 4; ++hb) agg[hb][r] *= inv;
  }

  // ---- FFN layer 1: relu(X @ W1 + b1), X staged through LDS ----
#pragma unroll
  for (int hb = 0; hb < 4; ++hb)
#pragma unroll
    for (int r = 0; r < 8; ++r)
      myl[(r + 8 * hi) * 68 + hb * 16 + rowM] = agg[hb][r];
  v2f xa[16];
#pragma unroll
  for (int k = 0; k < 16; ++k)
    xa[k] = *(const v2f*)(myl + rowM * 68 + k * 4 + 2 * hi);
  v8f h1[4];
#pragma unroll
  for (int hb = 0; hb < 4; ++hb) {
    v8f acc = {0.f, 0.f, 0.f, 0.f, 0.f, 0.f, 0.f, 0.f};
#pragma unroll
    for (int k = 0; k < 16; ++k) {
      v2f wb;
      wb[0] = W1[(size_t)(k * 4 + 2 * hi) * H_ + hb * 16 + rowM];
      wb[1] = W1[(size_t)(k * 4 + 2 * hi + 1) * H_ + hb * 16 + rowM];
      acc = wmma_f32(xa[k], wb, acc);
    }
    float bias = b1[hb * 16 + rowM];
#pragma unroll
    for (int r = 0; r < 8; ++r) h1[hb][r] = fmaxf(acc[r] + bias, 0.f);
  }

  // ---- FFN layer 2: H1 @ W2 + b2, residual add ----
#pragma unroll
  for (int hb = 0; hb < 4; ++hb)
#pragma unroll
    for (int r = 0; r < 8; ++r)
      myl[(r + 8 * hi) * 68 + hb * 16 + rowM] = h1[hb][r];
#pragma unroll
  for (int k = 0; k < 16; ++k)
    xa[k] = *(const v2f*)(myl + rowM * 68 + k * 4 + 2 * hi);
  const float* res_base = feat + (size_t)((b * T_ + t + 2) * N_) * H_;
  const int nodebase = tile * 16;
  v8f sv[4];
  float gam[4], bet[4];
#pragma unroll
  for (int hb = 0; hb < 4; ++hb) {
    v8f acc = {0.f, 0.f, 0.f, 0.f, 0.f, 0.f, 0.f, 0.f};
#pragma unroll
    for (int k = 0; k < 16; ++k) {
      v2f wb;
      wb[0] = W2[(size_t)(k * 4 + 2 * hi) * H_ + hb * 16 + rowM];
      wb[1] = W2[(size_t)(k * 4 + 2 * hi + 1) * H_ + hb * 16 + rowM];
      acc = wmma_f32(xa[k], wb, acc);
    }
    float bias = b2[hb * 16 + rowM];
    gam[hb] = gamma[hb * 16 + rowM];
    bet[hb] = beta[hb * 16 + rowM];
#pragma unroll
    for (int r = 0; r < 8; ++r) {
      int node = nodebase + r + 8 * hi;
      int nc = node < N_ ? node : N_ - 1;
      sv[hb][r] = acc[r] + bias + res_base[(size_t)nc * H_ + hb * 16 + rowM];
    }
  }

  // ---- LayerNorm over H (per matrix row: 4 reg-blocks x 16 half-lanes) ----
#pragma unroll
  for (int r = 0; r < 8; ++r) {
    float s1 = sv[0][r] + sv[1][r] + sv[2][r] + sv[3][r];
    float s2 = sv[0][r] * sv[0][r] + sv[1][r] * sv[1][r] +
               sv[2][r] * sv[2][r] + sv[3][r] * sv[3][r];
    s1 += __shfl_xor(s1, 1); s1 += __shfl_xor(s1, 2);
    s1 += __shfl_xor(s1, 4); s1 += __shfl_xor(s1, 8);
    s2 += __shfl_xor(s2, 1); s2 += __shfl_xor(s2, 2);
    s2 += __shfl_xor(s2, 4); s2 += __shfl_xor(s2, 8);
    float mu  = s1 * (1.f / 64.f);
    float var = s2 * (1.f / 64.f) - mu * mu;
    float inv = rsqrtf(var + 1e-5f);
#pragma unroll
    for (int hb = 0; hb < 4; ++hb)
      sv[hb][r] = (sv[hb][r] - mu) * inv * gam[hb] + bet[hb];
  }

  float* out_base = out + (size_t)((b * TW_ + t) * N_) * H_;
#pragma unroll
  for (int hb = 0; hb < 4; ++hb)
#pragma unroll
    for (int r = 0; r < 8; ++r) {
      int node = nodebase + r + 8 * hi;
      if (node < N_)
        out_base[(size_t)node * H_ + hb * 16 + rowM] = sv[hb][r];
    }
}

extern "C" void kernel_launch(void* const* d_in, const int* in_sizes, int n_in,
                              void* d_out, int out_size, void* d_ws,
                              size_t ws_size, hipStream_t stream) {
  (void)in_sizes; (void)n_in; (void)out_size; (void)ws_size;
  const float* feat = (const float*)d_in[0];
  const float* w    = (const float*)d_in[1];
  const float* W1   = (const float*)d_in[2];
  const float* b1   = (const float*)d_in[3];
  const float* W2   = (const float*)d_in[4];
  const float* b2   = (const float*)d_in[5];
  const float* gm   = (const float*)d_in[6];
  const float* bt   = (const float*)d_in[7];
  float* wfbuf = (float*)d_ws;  // B*T*N*H floats = 40.7 MB scratch
  float* outp  = (float*)d_out;

  const int nodes = B_ * T_ * N_;
  gnn_prep_kernel<<<(nodes + 255) / 256, 256, 0, stream>>>(feat, w, wfbuf);

  const int waves  = B_ * TW_ * NTILE_;  // 8320
  const int blocks = (waves + 7) / 8;    // 8 waves per 256-thread block
  gnn_main_kernel<<<blocks, 256, 0, stream>>>(feat, wfbuf, W1, b1, W2, b2, gm,
                                              bt, outp);
}